// FrontierAttention_50792283242750
// MI455X (gfx1250) — compile-verified
//
#include <hip/hip_runtime.h>
#include <hip/hip_bf16.h>
#include <math.h>

#define B_  4
#define N_  50000
#define C_  128
#define S_  64
#define H_  8
#define E_  800000
#define DH_ 16
#define BETA_ 1.0f
#define EPS_  1e-5f

typedef __attribute__((ext_vector_type(2))) float v2f;
typedef __attribute__((ext_vector_type(8))) float v8f;

// ---------------- K1: LayerNorm: h = LN(x) ----------------
// one wave (32 lanes) per row of 128 channels, float4 per lane
__global__ __launch_bounds__(256) void k_layernorm(const float* __restrict__ x,
                                                   const float* __restrict__ g,
                                                   const float* __restrict__ bt,
                                                   float* __restrict__ h) {
    int row  = blockIdx.x * 8 + (threadIdx.x >> 5);
    int lane = threadIdx.x & 31;
    if (row >= B_ * N_) return;
    const float4* xr = (const float4*)(x + (size_t)row * C_);
    float4 v = xr[lane];
    float s = v.x + v.y + v.z + v.w;
#pragma unroll
    for (int m = 16; m >= 1; m >>= 1) s += __shfl_xor(s, m, 32);
    float mean = s * (1.0f / C_);
    float dx = v.x - mean, dy = v.y - mean, dz = v.z - mean, dw = v.w - mean;
    float q = dx*dx + dy*dy + dz*dz + dw*dw;
#pragma unroll
    for (int m = 16; m >= 1; m >>= 1) q += __shfl_xor(q, m, 32);
    float rstd = rsqrtf(q * (1.0f / C_) + EPS_);
    float4 gg = ((const float4*)g)[lane];
    float4 bb = ((const float4*)bt)[lane];
    float4 o;
    o.x = dx * rstd * gg.x + bb.x;
    o.y = dy * rstd * gg.y + bb.y;
    o.z = dz * rstd * gg.z + bb.z;
    o.w = dw * rstd * gg.w + bb.w;
    ((float4*)(h + (size_t)row * C_))[lane] = o;
}

// ---------------- K2: SpMM scatter-add: h_local[b,r] += val*h[b,c] ----------
// one block per edge; threadIdx.y = batch, lanes cover 128 channels as float4
__global__ __launch_bounds__(128) void k_spmm(const float* __restrict__ h,
                                              const int* __restrict__ idx,
                                              const float* __restrict__ val,
                                              float* __restrict__ hl) {
    int e = blockIdx.x;
    int b = threadIdx.y;
    int lane = threadIdx.x;
    int r = idx[e];
    int c = idx[E_ + e];
    float w = val[e];
    float4 v = ((const float4*)(h + ((size_t)b * N_ + c) * C_))[lane];
    float* dst = hl + ((size_t)b * N_ + r) * C_ + lane * 4;
    atomicAdd(dst + 0, v.x * w);
    atomicAdd(dst + 1, v.y * w);
    atomicAdd(dst + 2, v.z * w);
    atomicAdd(dst + 3, v.w * w);
}

// ---------------- K3: logits + softmax -> weights, wsum ----------------
// logits = h@(Ws+beta*Wf)^T - h_local@(beta*Wf)^T + biases   (since hf = h - hl)
// Per wave: 16 rows x 64 cols via v_wmma_f32_16x16x4_f32; softmax in registers.
// Out-of-range rows load from a clamped (valid) pointer and compute garbage
// that is masked at the store / wsum stage -> no divergence in the hot loop.
__global__ __launch_bounds__(256) void k_logits_softmax(
    const float* __restrict__ h, const float* __restrict__ hl,
    const float* __restrict__ sw, const float* __restrict__ sb,
    const float* __restrict__ fw, const float* __restrict__ fb,
    float* __restrict__ wout, float* __restrict__ wsum) {
    __shared__ float WC[C_ * S_];   // transposed (k, s): Ws + beta*Wf
    __shared__ float WF[C_ * S_];   // transposed (k, s): beta*Wf
    int tid = threadIdx.x;
    for (int i = tid; i < C_ * S_; i += 256) {
        int s = i >> 7, k = i & 127;
        float f = BETA_ * fw[s * C_ + k];
        WC[k * S_ + s] = sw[s * C_ + k] + f;
        WF[k * S_ + s] = f;
    }
    __syncthreads();

    int b = blockIdx.y;
    int lane = tid & 31, wave = tid >> 5;
    int hi = lane >> 4, lo = lane & 15, koff = hi * 2;
    int rowbase = blockIdx.x * 128 + wave * 16;
    int arow = rowbase + lo;
    int crow = arow < N_ ? arow : 0;               // clamped, always valid
    const float* hr = h  + ((size_t)b * N_ + crow) * C_;
    const float* lr = hl + ((size_t)b * N_ + crow) * C_;

    v8f acc[4];
#pragma unroll
    for (int t = 0; t < 4; ++t)
#pragma unroll
        for (int v = 0; v < 8; ++v) acc[t][v] = 0.0f;

    for (int k0 = 0; k0 < C_; k0 += 4) {
        v2f Ah, An;
        Ah.x = hr[k0 + koff];
        Ah.y = hr[k0 + koff + 1];
        An.x = -lr[k0 + koff];                     // subtract the h_local term
        An.y = -lr[k0 + koff + 1];
#pragma unroll
        for (int t = 0; t < 4; ++t) {
            v2f Bc, Bf;
            Bc.x = WC[(k0 + koff)     * S_ + t * 16 + lo];
            Bc.y = WC[(k0 + koff + 1) * S_ + t * 16 + lo];
            Bf.x = WF[(k0 + koff)     * S_ + t * 16 + lo];
            Bf.y = WF[(k0 + koff + 1) * S_ + t * 16 + lo];
            acc[t] = __builtin_amdgcn_wmma_f32_16x16x4_f32(false, Ah, false, Bc,
                                                           (short)0, acc[t], false, false);
            acc[t] = __builtin_amdgcn_wmma_f32_16x16x4_f32(false, An, false, Bf,
                                                           (short)0, acc[t], false, false);
        }
    }

    float bias[4];
#pragma unroll
    for (int t = 0; t < 4; ++t) bias[t] = sb[t * 16 + lo] + BETA_ * fb[t * 16 + lo];

    // softmax per D-row: row = rowbase + hi*8 + v, held by 16 lanes (4 cols each)
#pragma unroll
    for (int v = 0; v < 8; ++v) {
        float x0 = acc[0][v] + bias[0], x1 = acc[1][v] + bias[1];
        float x2 = acc[2][v] + bias[2], x3 = acc[3][v] + bias[3];
        float mx = fmaxf(fmaxf(x0, x1), fmaxf(x2, x3));
#pragma unroll
        for (int m = 1; m <= 8; m <<= 1) mx = fmaxf(mx, __shfl_xor(mx, m, 32));
        float e0 = __expf(x0 - mx), e1 = __expf(x1 - mx);
        float e2 = __expf(x2 - mx), e3 = __expf(x3 - mx);
        float sm = e0 + e1 + e2 + e3;
#pragma unroll
        for (int m = 1; m <= 8; m <<= 1) sm += __shfl_xor(sm, m, 32);
        float inv = 1.0f / sm;
        acc[0][v] = e0 * inv; acc[1][v] = e1 * inv;
        acc[2][v] = e2 * inv; acc[3][v] = e3 * inv;
    }

#pragma unroll
    for (int t = 0; t < 4; ++t) {
        float cs = 0.f;
#pragma unroll
        for (int v = 0; v < 8; ++v) {
            int rowD = rowbase + hi * 8 + v;
            if (rowD < N_) {
                float w = acc[t][v];
                wout[((size_t)b * N_ + rowD) * S_ + t * 16 + lo] = w;
                cs += w;
            }
        }
        cs += __shfl_xor(cs, 16, 32);   // lane l and l^16 hold the same column
        if (lane < 16) atomicAdd(&wsum[b * S_ + t * 16 + lane], cs);
    }
}

// ---------------- K4: slices_acc += weights^T @ h (split-K over N) ----------
// Requires chunk % 4 == 0 and N_ % 4 == 0 so every k-step is fully in-bounds.
__global__ __launch_bounds__(256) void k_slices(const float* __restrict__ wts,
                                                const float* __restrict__ h,
                                                float* __restrict__ sacc, int chunk) {
    int b  = blockIdx.z;
    int s0 = blockIdx.y * 16;
    int n0 = blockIdx.x * chunk;
    int nend = n0 + chunk; if (nend > N_) nend = N_;
    int tid = threadIdx.x, lane = tid & 31, wave = tid >> 5;
    int c0 = wave * 16;
    int hi = lane >> 4, lo = lane & 15, koff = hi * 2;

    v8f acc;
#pragma unroll
    for (int v = 0; v < 8; ++v) acc[v] = 0.0f;

    const float* wp = wts + (size_t)b * N_ * S_;
    const float* hp = h   + (size_t)b * N_ * C_;
    for (int k0 = n0; k0 < nend; k0 += 4) {
        int na = k0 + koff;                        // na, na+1 < nend guaranteed
        v2f A, Bv;
        A.x  = wp[(size_t)na * S_ + s0 + lo];
        A.y  = wp[(size_t)(na + 1) * S_ + s0 + lo];
        Bv.x = hp[(size_t)na * C_ + c0 + lo];
        Bv.y = hp[(size_t)(na + 1) * C_ + c0 + lo];
        acc = __builtin_amdgcn_wmma_f32_16x16x4_f32(false, A, false, Bv,
                                                    (short)0, acc, false, false);
    }
#pragma unroll
    for (int v = 0; v < 8; ++v) {
        int s = s0 + hi * 8 + v;
        atomicAdd(&sacc[((size_t)b * S_ + s) * C_ + c0 + lo], acc[v]);
    }
}

// ---------------- Small kernels: normalize / qkv / attention / out-proj -----
__global__ void k_norm_slices(const float* __restrict__ sacc,
                              const float* __restrict__ wsum,
                              float* __restrict__ slices) {
    int i = blockIdx.x * 256 + threadIdx.x;
    if (i >= B_ * S_ * C_) return;
    slices[i] = sacc[i] / fmaxf(wsum[i / C_], 1e-8f);
}

__global__ void k_qkv(const float* __restrict__ slices, const float* __restrict__ iw,
                      const float* __restrict__ ib, float* __restrict__ qkv) {
    int i = blockIdx.x * 256 + threadIdx.x;
    if (i >= B_ * S_ * 3 * C_) return;
    int j  = i % (3 * C_);
    int bs = i / (3 * C_);
    const float* srow = slices + (size_t)bs * C_;
    const float* wrow = iw + (size_t)j * C_;
    float a = ib[j];
#pragma unroll 4
    for (int c = 0; c < C_; ++c) a += srow[c] * wrow[c];
    qkv[i] = a;
}

__global__ __launch_bounds__(64) void k_attn(const float* __restrict__ qkv,
                                             float* __restrict__ o) {
    int b = blockIdx.x / H_, hh = blockIdx.x % H_;
    int r = threadIdx.x;                       // query row 0..63
    const float* base = qkv + (size_t)b * S_ * 3 * C_;
    float sc[S_];
    float mx = -1e30f;
    for (int kk = 0; kk < S_; ++kk) {
        float d = 0.f;
#pragma unroll
        for (int dd = 0; dd < DH_; ++dd)
            d += base[r * 3 * C_ + hh * DH_ + dd] *
                 base[kk * 3 * C_ + C_ + hh * DH_ + dd];
        d *= 0.25f;                            // 1/sqrt(DH)
        sc[kk] = d;
        mx = fmaxf(mx, d);
    }
    float sum = 0.f;
    for (int kk = 0; kk < S_; ++kk) { sc[kk] = __expf(sc[kk] - mx); sum += sc[kk]; }
    float inv = 1.0f / sum;
#pragma unroll
    for (int dd = 0; dd < DH_; ++dd) {
        float a = 0.f;
        for (int kk = 0; kk < S_; ++kk)
            a += sc[kk] * base[kk * 3 * C_ + 2 * C_ + hh * DH_ + dd];
        o[((size_t)b * S_ + r) * C_ + hh * DH_ + dd] = a * inv;
    }
}

__global__ void k_outproj(const float* __restrict__ o, const float* __restrict__ ow,
                          const float* __restrict__ ob, float* __restrict__ so) {
    int i = blockIdx.x * 256 + threadIdx.x;
    if (i >= B_ * S_ * C_) return;
    int c  = i % C_;
    int bs = i / C_;
    const float* orow = o + (size_t)bs * C_;
    const float* wrow = ow + (size_t)c * C_;
    float a = ob[c];
#pragma unroll 4
    for (int k = 0; k < C_; ++k) a += orow[k] * wrow[k];
    so[i] = a;
}

// ---------------- K5: out = weights @ slices_out ----------------
// Unconditional A loads from clamped rows; stores guarded.
__global__ __launch_bounds__(256) void k_out(const float* __restrict__ wts,
                                             const float* __restrict__ so,
                                             float* __restrict__ out) {
    __shared__ float lso[S_ * C_];   // 32KB: slices_out for this batch
    int b = blockIdx.y;
    int tid = threadIdx.x;
    for (int i = tid; i < S_ * C_; i += 256) lso[i] = so[(size_t)b * S_ * C_ + i];
    __syncthreads();

    int lane = tid & 31, wave = tid >> 5;
    int hi = lane >> 4, lo = lane & 15, koff = hi * 2;
    int rowbase = blockIdx.x * 128 + wave * 16;
    int arow = rowbase + lo;
    int crow = arow < N_ ? arow : 0;               // clamped, always valid
    const float* wr = wts + ((size_t)b * N_ + crow) * S_;

    v8f acc[8];
#pragma unroll
    for (int t = 0; t < 8; ++t)
#pragma unroll
        for (int v = 0; v < 8; ++v) acc[t][v] = 0.0f;

    for (int k0 = 0; k0 < S_; k0 += 4) {
        v2f A;
        A.x = wr[k0 + koff];
        A.y = wr[k0 + koff + 1];
#pragma unroll
        for (int t = 0; t < 8; ++t) {
            v2f Bv;
            Bv.x = lso[(k0 + koff)     * C_ + t * 16 + lo];
            Bv.y = lso[(k0 + koff + 1) * C_ + t * 16 + lo];
            acc[t] = __builtin_amdgcn_wmma_f32_16x16x4_f32(false, A, false, Bv,
                                                           (short)0, acc[t], false, false);
        }
    }
#pragma unroll
    for (int t = 0; t < 8; ++t)
#pragma unroll
        for (int v = 0; v < 8; ++v) {
            int rowD = rowbase + hi * 8 + v;
            if (rowD < N_)
                out[((size_t)b * N_ + rowD) * C_ + t * 16 + lo] = acc[t][v];
        }
}

extern "C" void kernel_launch(void* const* d_in, const int* in_sizes, int n_in,
                              void* d_out, int out_size, void* d_ws, size_t ws_size,
                              hipStream_t stream) {
    const float* x          = (const float*)d_in[0];
    const float* adj_values = (const float*)d_in[1];
    const float* ln_g       = (const float*)d_in[2];
    const float* ln_bv      = (const float*)d_in[3];
    const float* slice_w    = (const float*)d_in[4];
    const float* slice_b    = (const float*)d_in[5];
    const float* front_w    = (const float*)d_in[6];
    const float* front_b    = (const float*)d_in[7];
    const float* in_w       = (const float*)d_in[8];
    const float* in_b       = (const float*)d_in[9];
    const float* out_w      = (const float*)d_in[10];
    const float* out_b      = (const float*)d_in[11];
    const int*   adj_idx    = (const int*)d_in[12];

    float* out  = (float*)d_out;                       // (B,N,C)
    float* wout = out + (size_t)B_ * N_ * C_;          // (B,N,S) weights output

    float* h      = (float*)d_ws;                      // (B,N,C)
    float* hl     = h + (size_t)B_ * N_ * C_;          // (B,N,C)
    float* sacc   = hl + (size_t)B_ * N_ * C_;         // (B,S,C)
    float* wsum   = sacc + (size_t)B_ * S_ * C_;       // (B,S)
    float* slices = wsum + B_ * S_;                    // (B,S,C)
    float* qkv    = slices + (size_t)B_ * S_ * C_;     // (B,S,3C)
    float* atno   = qkv + (size_t)B_ * S_ * 3 * C_;    // (B,S,C)
    float* so     = atno + (size_t)B_ * S_ * C_;       // (B,S,C)

    k_layernorm<<<dim3((B_ * N_ + 7) / 8), 256, 0, stream>>>(x, ln_g, ln_bv, h);

    hipMemsetAsync(hl, 0, (size_t)B_ * N_ * C_ * sizeof(float), stream);
    k_spmm<<<dim3(E_), dim3(32, 4), 0, stream>>>(h, adj_idx, adj_values, hl);

    hipMemsetAsync(wsum, 0, B_ * S_ * sizeof(float), stream);
    hipMemsetAsync(sacc, 0, (size_t)B_ * S_ * C_ * sizeof(float), stream);
    k_logits_softmax<<<dim3((N_ + 127) / 128, B_), 256, 0, stream>>>(
        h, hl, slice_w, slice_b, front_w, front_b, wout, wsum);

    const int chunk = 1024;                            // multiple of 4
    k_slices<<<dim3((N_ + chunk - 1) / chunk, S_ / 16, B_), 256, 0, stream>>>(
        wout, h, sacc, chunk);

    k_norm_slices<<<dim3((B_ * S_ * C_ + 255) / 256), 256, 0, stream>>>(sacc, wsum, slices);
    k_qkv<<<dim3((B_ * S_ * 3 * C_ + 255) / 256), 256, 0, stream>>>(slices, in_w, in_b, qkv);
    k_attn<<<dim3(B_ * H_), 64, 0, stream>>>(qkv, atno);
    k_outproj<<<dim3((B_ * S_ * C_ + 255) / 256), 256, 0, stream>>>(atno, out_w, out_b, so);

    k_out<<<dim3((N_ + 127) / 128, B_), 256, 0, stream>>>(wout, so, out);
}